// MultiCGLayer_20091857010911
// MI455X (gfx1250) — compile-verified
//
#include <hip/hip_runtime.h>

typedef __attribute__((ext_vector_type(2))) float v2f;
typedef __attribute__((ext_vector_type(8))) float v8f;

#define N_COMBOS 15
#define KP 164           // padded u dimension (162 -> 164, multiple of 4)
#define PN 112           // padded raw-message dimension (102 -> 112, 7 N-tiles)
#define NTILES 7
#define WAVES 2          // waves (16-edge tiles) per workgroup
#define FEAT 18

// compile-time tables (constexpr so the compiler can fold every index / guard)
constexpr int cLE[N_COMBOS]     = {0,0,0,1,1,1,1,1,1,2,2,2,2,2,2};
constexpr int cLN[N_COMBOS]     = {0,1,2,0,1,1,1,2,2,0,1,1,2,2,2};
constexpr int cLO[N_COMBOS]     = {0,1,2,1,0,1,2,1,2,2,1,2,0,1,2};
constexpr int OUT_OFF[N_COMBOS] = {0,2,8,18,24,26,32,42,48,58,68,74,84,86,92};
constexpr int U_OFF[9]          = {0,2,8,18,24,42,72,82,112};
// per-N-tile active K-tile range (K tiles of 4 rows), from block-sparsity of M
constexpr int KLO[NTILES]       = {0,2,6,10,18,20,28};
constexpr int KHI[NTILES]       = {4,10,17,20,27,40,40};
constexpr int N_BY_LO[3]        = {3,6,6};
constexpr int C_BY_LO[3][6]     = {{0,4,12,0,0,0},{1,3,5,7,10,13},{2,6,8,9,11,14}};

// ---------------------------------------------------------------------------
// Build the edge-independent CG matrix M [KP x PN] from the 15 cg tensors.
// u-row index: U_OFF(le,ln) + (a*dn+b)*2 + ch ; col: OUT_OFF(c) + ch*do + k
// ---------------------------------------------------------------------------
__global__ void build_M_kernel(const float* cg0, const float* cg1, const float* cg2,
                               const float* cg3, const float* cg4, const float* cg5,
                               const float* cg6, const float* cg7, const float* cg8,
                               const float* cg9, const float* cg10, const float* cg11,
                               const float* cg12, const float* cg13, const float* cg14,
                               float* M) {
    const float* cgs[N_COMBOS] = {cg0,cg1,cg2,cg3,cg4,cg5,cg6,cg7,cg8,cg9,cg10,cg11,cg12,cg13,cg14};
    int tid = threadIdx.x;
    for (int i = tid; i < KP * PN; i += blockDim.x) M[i] = 0.0f;
    __syncthreads();
    if (tid < N_COMBOS) {
        int c = tid;
        int le = cLE[c], ln = cLN[c], lo = cLO[c];
        int de = 2 * le + 1, dn = 2 * ln + 1, dO = 2 * lo + 1;
        int uo = U_OFF[le * 3 + ln];
        int po = OUT_OFF[c];
        const float* cg = cgs[c];
        for (int a = 0; a < de; ++a)
            for (int b = 0; b < dn; ++b)
                for (int k = 0; k < dO; ++k) {
                    float v = cg[(a * dn + b) * dO + k];
                    for (int ch = 0; ch < 2; ++ch)
                        M[(uo + (a * dn + b) * 2 + ch) * PN + po + ch * dO + k] = v;
                }
    }
}

__global__ void zero_kernel(float* p, long n) {
    long i = (long)blockIdx.x * blockDim.x + threadIdx.x;
    if (i < n) p[i] = 0.0f;
}

// ---------------------------------------------------------------------------
// part-specialized helpers: all indices compile-time constant after unrolling
// ---------------------------------------------------------------------------
template <int PART>
__device__ __forceinline__ void gate_block(const float* __restrict__ W1,
                                           const float* __restrict__ b1,
                                           const float* __restrict__ W2,
                                           const float* __restrict__ b2,
                                           float x0, float x1,
                                           float* __restrict__ gRow) {
#pragma unroll
    for (int t = 0; t < 30; ++t) {
        const int g = PART * 30 + t;                // compile-time per iteration
        const int bi = ((g >> 2) * 2 + ((g >> 1) & 1)) * 2 + (g & 1);
        float acc = b2[bi];
#pragma unroll
        for (int h = 0; h < 4; ++h) {
            float hh = x0 * W1[bi * 8 + h * 2] + x1 * W1[bi * 8 + h * 2 + 1] + b1[bi * 4 + h];
            hh = hh > 0.0f ? hh : 0.0f;
            acc += hh * W2[bi * 4 + h];
        }
        gRow[g] = tanhf(acc);
    }
}

template <int PART>
__device__ __forceinline__ void u_stage(const float* __restrict__ shv,
                                        const float* __restrict__ f,
                                        float* __restrict__ uRow) {
    int u = 0;
#pragma unroll
    for (int le = 0; le < 3; ++le) {
        const int de = 2 * le + 1;
        const int shb = (le == 0) ? 0 : (le == 1 ? 1 : 4);
#pragma unroll
        for (int ln = 0; ln < 3; ++ln) {
            const int dn = 2 * ln + 1;
            const int fb = (ln == 0) ? 0 : (ln == 1 ? 2 : 8);
#pragma unroll
            for (int a = 0; a < de; ++a)
#pragma unroll
                for (int b = 0; b < dn; ++b)
#pragma unroll
                    for (int ch = 0; ch < 2; ++ch) {
                        if ((PART == 0) ? (u < 81) : (u >= 81))
                            uRow[u] = shv[shb + a] * f[fb + ch * dn + b];
                        ++u;
                    }
        }
    }
    if (PART == 1) { uRow[162] = 0.0f; uRow[163] = 0.0f; }
}

template <int S>
__device__ __forceinline__ float slot_val(const float* __restrict__ gRow,
                                          const float* __restrict__ rRow) {
    constexpr int lo = (S < 2) ? 0 : ((S < 8) ? 1 : 2);
    constexpr int o  = (S < 2) ? S : ((S < 8) ? ((S - 2) / 3) : ((S - 8) / 5));
    constexpr int k  = (S < 2) ? 0 : ((S < 8) ? ((S - 2) % 3) : ((S - 8) % 5));
    constexpr int dO = 2 * lo + 1;
    float v = 0.0f;
#pragma unroll
    for (int q = 0; q < N_BY_LO[lo]; ++q) {
        constexpr const int* lst = C_BY_LO[lo];
        const int c = lst[q];                       // folds: lo, q compile-time
        const int base = OUT_OFF[c];
        v += gRow[c * 4 + o]     * rRow[base + k] +
             gRow[c * 4 + 2 + o] * rRow[base + dO + k];
    }
    return v;
}

// ---------------------------------------------------------------------------
// One message-passing layer. Each wave processes a 16-edge tile:
//   phase 1: gather + invariants + gate MLPs (VALU), stage u tile in LDS
//   phase 2: raw = u @ M via V_WMMA_F32_16X16X4_F32 (81 WMMAs, sparse K folded)
//   phase 3: gate-mix channels, scatter-add into out features (f32 atomics)
// ---------------------------------------------------------------------------
__global__ void __launch_bounds__(WAVES * 32)
layer_kernel(const float* __restrict__ feats, const int* __restrict__ eidx,
             const float* __restrict__ sh0, const float* __restrict__ sh1,
             const float* __restrict__ sh2, const float* __restrict__ dist,
             const float* __restrict__ Mg,
             const float* __restrict__ W1, const float* __restrict__ b1,
             const float* __restrict__ W2, const float* __restrict__ b2,
             float* __restrict__ outF, int E) {
    __shared__ float sU[WAVES][16 * KP];
    __shared__ float sRaw[WAVES][16 * PN];
    __shared__ float sGate[WAVES][16 * 60];

    const int lane = threadIdx.x & 31;
    const int w = threadIdx.x >> 5;
    const int tile = blockIdx.x * WAVES + w;
    const int eSub = lane & 15;
    const int part = lane >> 4;           // two lanes cooperate per edge
    const long e = (long)tile * 16 + eSub;
    const bool valid = (e < (long)E);

    // ---------------- phase 1: gather, invariants, gates, u staging ---------
    float f[FEAT];
    float shv[9];
    float x0 = 0.0f, x1 = 0.0f;
    int tgt = 0;
    if (valid) {
        const float* fp = &feats[(long)eidx[e] * FEAT];
        tgt = eidx[(long)E + e];
#pragma unroll
        for (int j = 0; j < FEAT; ++j) f[j] = fp[j];
        shv[0] = sh0[e];
#pragma unroll
        for (int j = 0; j < 3; ++j) shv[1 + j] = sh1[e * 3 + j];
#pragma unroll
        for (int j = 0; j < 5; ++j) shv[4 + j] = sh2[e * 5 + j];
        x0 = dist[e];
        float inv0 = 0.5f * (f[0] + f[1]);
        float n1 = 0.0f, n2 = 0.0f;
#pragma unroll
        for (int j = 0; j < 3; ++j) { float m = 0.5f * (f[2 + j] + f[5 + j]); n1 += m * m; }
#pragma unroll
        for (int j = 0; j < 5; ++j) { float m = 0.5f * (f[8 + j] + f[13 + j]); n2 += m * m; }
        x1 = (inv0 + sqrtf(n1) + sqrtf(n2)) * (1.0f / 3.0f);
    } else {
#pragma unroll
        for (int j = 0; j < FEAT; ++j) f[j] = 0.0f;
#pragma unroll
        for (int j = 0; j < 9; ++j) shv[j] = 0.0f;
    }

    float* gRow = &sGate[w][eSub * 60];
    float* uRow = &sU[w][eSub * KP];
    if (part == 0) {
        gate_block<0>(W1, b1, W2, b2, x0, x1, gRow);
        u_stage<0>(shv, f, uRow);
    } else {
        gate_block<1>(W1, b1, W2, b2, x0, x1, gRow);
        u_stage<1>(shv, f, uRow);
    }
    __syncthreads();

    // ---------------- phase 2: raw = u @ M via f32 WMMA ---------------------
    {
        const int col16 = lane & 15;
        const int kofs = (lane < 16) ? 0 : 2;   // ISA A/B fragment K split by lane half
        const float* uA = &sU[w][col16 * KP];
#pragma unroll
        for (int n = 0; n < NTILES; ++n) {
            v8f acc = {0.f, 0.f, 0.f, 0.f, 0.f, 0.f, 0.f, 0.f};
            const int col = col16 + n * 16;
#pragma unroll
            for (int kt = KLO[n]; kt <= KHI[n]; ++kt) {   // bounds fold: constexpr
                int k0 = kt * 4 + kofs;
                v2f a, b;
                a.x = uA[k0];
                a.y = uA[k0 + 1];
                b.x = Mg[k0 * PN + col];
                b.y = Mg[(k0 + 1) * PN + col];
                acc = __builtin_amdgcn_wmma_f32_16x16x4_f32(
                    false, a, false, b, (short)0, acc, false, false);
            }
            const int rbase = (lane < 16) ? 0 : 8;  // C/D layout: VGPR j -> row j (+8)
#pragma unroll
            for (int j = 0; j < 8; ++j)
                sRaw[w][(rbase + j) * PN + col] = acc[j];
        }
    }
    __syncthreads();

    // ---------------- phase 3: gate-mix + scatter-add -----------------------
    {
        const float* rRow = &sRaw[w][eSub * PN];
        float* outBase = &outF[(long)tgt * FEAT];
#define DO_SLOT(S)                                                     \
        { float v = slot_val<S>(gRow, rRow);                           \
          if (valid) atomicAdd(outBase + (S), v); }
        if (part == 0) {
            DO_SLOT(0) DO_SLOT(1) DO_SLOT(2) DO_SLOT(3) DO_SLOT(4)
            DO_SLOT(5) DO_SLOT(6) DO_SLOT(7) DO_SLOT(8)
        } else {
            DO_SLOT(9) DO_SLOT(10) DO_SLOT(11) DO_SLOT(12) DO_SLOT(13)
            DO_SLOT(14) DO_SLOT(15) DO_SLOT(16) DO_SLOT(17)
        }
#undef DO_SLOT
    }
}

// ---------------------------------------------------------------------------
extern "C" void kernel_launch(void* const* d_in, const int* in_sizes, int n_in,
                              void* d_out, int out_size, void* d_ws, size_t ws_size,
                              hipStream_t stream) {
    const float* node = (const float*)d_in[0];
    const int* eidx   = (const int*)d_in[1];
    const float* sh0  = (const float*)d_in[2];
    const float* sh1  = (const float*)d_in[3];
    const float* sh2  = (const float*)d_in[4];
    const float* dist = (const float*)d_in[5];
    const float* cg[N_COMBOS];
    for (int i = 0; i < N_COMBOS; ++i) cg[i] = (const float*)d_in[6 + i];
    const float* W1_0 = (const float*)d_in[21];
    const float* b1_0 = (const float*)d_in[22];
    const float* W2_0 = (const float*)d_in[23];
    const float* b2_0 = (const float*)d_in[24];
    const float* W1_1 = (const float*)d_in[25];
    const float* b1_1 = (const float*)d_in[26];
    const float* W2_1 = (const float*)d_in[27];
    const float* b2_1 = (const float*)d_in[28];

    const int E = in_sizes[1] / 2;
    const int N = in_sizes[0] / FEAT;

    float* Mg     = (float*)d_ws;                          // KP*PN*4 = 73,472 B
    float* feats1 = (float*)((char*)d_ws + 131072);        // N*18*4 = 3.6 MB

    build_M_kernel<<<1, 256, 0, stream>>>(cg[0], cg[1], cg[2], cg[3], cg[4], cg[5],
                                          cg[6], cg[7], cg[8], cg[9], cg[10], cg[11],
                                          cg[12], cg[13], cg[14], Mg);

    long n1 = (long)N * FEAT;
    zero_kernel<<<(int)((n1 + 255) / 256), 256, 0, stream>>>(feats1, n1);
    zero_kernel<<<(out_size + 255) / 256, 256, 0, stream>>>((float*)d_out, (long)out_size);

    int grid = (E + WAVES * 16 - 1) / (WAVES * 16);
    layer_kernel<<<grid, WAVES * 32, 0, stream>>>(node, eidx, sh0, sh1, sh2, dist, Mg,
                                                  W1_0, b1_0, W2_0, b2_0, feats1, E);
    layer_kernel<<<grid, WAVES * 32, 0, stream>>>(feats1, eidx, sh0, sh1, sh2, dist, Mg,
                                                  W1_1, b1_1, W2_1, b2_1, (float*)d_out, E);
}